// HPCGCN_23527830847932
// MI455X (gfx1250) — compile-verified
//
#include <hip/hip_runtime.h>

typedef __attribute__((ext_vector_type(2))) float v2f;
typedef __attribute__((ext_vector_type(8))) float v8f;

#define N_NODES 100000
#define N_EDGES 1600000
#define IN_DIM  128
#define HID_DIM 128
#define OUT_DIM 16
#define LDS_STRIDE 132   // 128 + 4 pad to dodge bank conflicts

// ---------------- degree / normalization ----------------
__global__ void edge_deg_kernel(const int* __restrict__ dst, float* __restrict__ deg) {
    int i = blockIdx.x * blockDim.x + threadIdx.x;
    if (i < N_EDGES) atomicAdd(&deg[dst[i]], 1.0f);
}

__global__ void dinv_kernel(float* __restrict__ deg) {
    int i = blockIdx.x * blockDim.x + threadIdx.x;
    if (i < N_NODES) deg[i] = rsqrtf(deg[i] + 1.0f);  // +1 self-loop
}

// ---------------- WMMA f32 GEMM: H = X(16-row strip) @ W [+ bias] ----------------
// Block: 32 * (KOUT/16) threads; wave w computes columns [16w, 16w+16).
// grid.x = N_NODES/16. K is fixed at 128 (IN_DIM == HID_DIM).
template <int KOUT, bool WITH_BIAS>
__global__ void gemm_wmma_kernel(const float* __restrict__ X,
                                 const float* __restrict__ W,
                                 const float* __restrict__ bias,
                                 float* __restrict__ H) {
    __shared__ float xs[16 * LDS_STRIDE];
    const int tid  = threadIdx.x;
    const int lane = tid & 31;
    const int wv   = tid >> 5;            // wave id -> column tile
    const int row0 = blockIdx.x * 16;

    // Stage the 16x128 X strip into LDS (float4), shared by all waves.
    for (int i = tid; i < 16 * (IN_DIM / 4); i += blockDim.x) {
        const int r  = i >> 5;            // / 32 float4s per row
        const int c4 = i & 31;
        const float4 v = ((const float4*)(X + (size_t)(row0 + r) * IN_DIM))[c4];
        float* p = &xs[r * LDS_STRIDE + c4 * 4];
        p[0] = v.x; p[1] = v.y; p[2] = v.z; p[3] = v.w;
    }
    __syncthreads();

    const int m    = lane & 15;           // row within tile / col within tile
    const int half = lane >> 4;           // 0: K pair {0,1}, 1: K pair {2,3}
    const int col  = wv * 16 + m;

    v8f c = {};
    #pragma unroll 8
    for (int kk = 0; kk < IN_DIM; kk += 4) {
        const int kb = kk + 2 * half;
        v2f a, b;
        // A 16x4 fragment: VGPR j holds K = j + 2*(lane/16), M = lane%16
        a.x = xs[m * LDS_STRIDE + kb];
        a.y = xs[m * LDS_STRIDE + kb + 1];
        // B 4x16 fragment: VGPR j holds K = j + 2*(lane/16), N = lane%16
        b.x = W[(size_t)(kb    ) * KOUT + col];
        b.y = W[(size_t)(kb + 1) * KOUT + col];
        // (neg_a, A, neg_b, B, c_mod, C, reuse_a, reuse_b)
        c = __builtin_amdgcn_wmma_f32_16x16x4_f32(false, a, false, b,
                                                  (short)0, c, false, false);
    }

    const float bv = WITH_BIAS ? bias[col] : 0.0f;
    #pragma unroll
    for (int v = 0; v < 8; ++v) {
        const int row = row0 + v + 8 * half;   // C layout: VGPR v -> M = v (+8 for hi lanes)
        H[(size_t)row * KOUT + col] = c[v] + bv;
    }
}

// ---------------- edge scatter: agg[dst] += h[src] * dinv[src]*dinv[dst] ----------------
// One wave per edge; each lane moves 4 contiguous floats (float4 gather + 4 f32 atomics).
__global__ void scatter_kernel(const int* __restrict__ src, const int* __restrict__ dst,
                               const float* __restrict__ dinv,
                               const float* __restrict__ h,
                               float* __restrict__ agg) {
    const long long gid = (long long)blockIdx.x * blockDim.x + threadIdx.x;
    const long long e   = gid >> 5;
    const int lane      = (int)(gid & 31);
    if (e >= N_EDGES) return;
    const int s = src[e], d = dst[e];
    const float coef = dinv[s] * dinv[d];
    const float4 v = ((const float4*)(h + (size_t)s * HID_DIM))[lane];
    float* out = agg + (size_t)d * HID_DIM + lane * 4;
    atomicAdd(out + 0, v.x * coef);
    atomicAdd(out + 1, v.y * coef);
    atomicAdd(out + 2, v.z * coef);
    atomicAdd(out + 3, v.w * coef);
}

// ---------------- self-loop + bias + ReLU (in place on agg) ----------------
__global__ void self_relu_kernel(const float* __restrict__ h,
                                 const float* __restrict__ dinv,
                                 const float* __restrict__ bias,
                                 float* __restrict__ agg) {
    const size_t idx = (size_t)blockIdx.x * blockDim.x + threadIdx.x;
    if (idx >= (size_t)N_NODES * HID_DIM) return;
    const int i = (int)(idx >> 7);        // / 128
    const int j = (int)(idx & 127);
    const float di = dinv[i];
    const float v = agg[idx] + h[idx] * (di * di) + bias[j];
    agg[idx] = v > 0.0f ? v : 0.0f;
}

// ---------------- launch ----------------
extern "C" void kernel_launch(void* const* d_in, const int* in_sizes, int n_in,
                              void* d_out, int out_size, void* d_ws, size_t ws_size,
                              hipStream_t stream) {
    const float* x  = (const float*)d_in[0];
    const int*   ei = (const int*)d_in[1];
    const float* W1 = (const float*)d_in[2];
    const float* b1 = (const float*)d_in[3];
    const float* W2 = (const float*)d_in[4];
    const float* b2 = (const float*)d_in[5];
    const float* Wf = (const float*)d_in[6];
    const float* bf = (const float*)d_in[7];
    float* out = (float*)d_out;

    const int* src = ei;            // edge_index[0]
    const int* dst = ei + N_EDGES;  // edge_index[1]

    float* dinv = (float*)d_ws;                                   // N floats
    float* bufA = dinv + ((N_NODES + 127) & ~127);                // N*128
    float* bufB = bufA + (size_t)N_NODES * HID_DIM;               // N*128

    const size_t hbytes = (size_t)N_NODES * HID_DIM * sizeof(float);
    const int elem_blocks = (int)(((size_t)N_NODES * HID_DIM + 255) / 256);
    const int scat_blocks = (int)(((long long)N_EDGES * 32 + 255) / 256);

    // degrees -> dinv
    hipMemsetAsync(dinv, 0, N_NODES * sizeof(float), stream);
    edge_deg_kernel<<<(N_EDGES + 255) / 256, 256, 0, stream>>>(dst, dinv);
    dinv_kernel<<<(N_NODES + 255) / 256, 256, 0, stream>>>(dinv);

    // layer 1
    gemm_wmma_kernel<HID_DIM, false><<<N_NODES / 16, 256, 0, stream>>>(x, W1, nullptr, bufA);
    hipMemsetAsync(bufB, 0, hbytes, stream);
    scatter_kernel<<<scat_blocks, 256, 0, stream>>>(src, dst, dinv, bufA, bufB);
    self_relu_kernel<<<elem_blocks, 256, 0, stream>>>(bufA, dinv, b1, bufB);   // bufB = h1

    // layer 2
    gemm_wmma_kernel<HID_DIM, false><<<N_NODES / 16, 256, 0, stream>>>(bufB, W2, nullptr, bufA);
    hipMemsetAsync(bufB, 0, hbytes, stream);
    scatter_kernel<<<scat_blocks, 256, 0, stream>>>(src, dst, dinv, bufA, bufB);
    self_relu_kernel<<<elem_blocks, 256, 0, stream>>>(bufA, dinv, b2, bufB);   // bufB = h2

    // head: out = h2 @ Wf + bf
    gemm_wmma_kernel<OUT_DIM, true><<<N_NODES / 16, 32, 0, stream>>>(bufB, Wf, bf, out);
}